// AdaFocalLoss_88098369175613
// MI455X (gfx1250) — compile-verified
//
#include <hip/hip_runtime.h>
#include <hip/hip_bf16.h>
#include <math.h>

// AdaFocal loss for (N=65536, C=1000) fp32 logits on gfx1250 (wave32).
// Memory-bound: stream 262 MB once at HBM rate (~11.3 us floor @ 23.3 TB/s).
// One row per wave32, whole row held in 32 VGPRs (8 x float4), two register
// passes (max, sum-exp), 5-step wave32 butterflies, one fp32 atomic / 8 rows.

#define NBINS 15
#define ROW_C 1000          // classes per row
#define CHUNKS 250          // float4 chunks per row (1000/4)
#define WAVES_PER_BLOCK 8
#define EPSF 1e-20f

__global__ __launch_bounds__(WAVES_PER_BLOCK * 32) void adafocal_kernel(
    const float* __restrict__ inp,     // [N, 1000]
    const int*   __restrict__ tgt,     // [N]
    const float* __restrict__ gammas,  // [15]
    float*       __restrict__ out,     // [1]
    int nrows)
{
    __shared__ float wave_loss[WAVES_PER_BLOCK];

    const int lane = threadIdx.x & 31;
    const int wave = threadIdx.x >> 5;
    const int row  = blockIdx.x * WAVES_PER_BLOCK + wave;

    float lossw = 0.0f;

    if (row < nrows) {
        const float4* rp = (const float4*)(inp + (size_t)row * ROW_C);

        // Load the whole row into registers: 8 x global_load_b128 per lane,
        // lanes 16B apart -> fully coalesced 512B per instruction per wave.
        float4 v[8];
        #pragma unroll
        for (int k = 0; k < 7; ++k) {
            v[k] = rp[lane + 32 * k];
        }
        {
            const int idx = lane + 224;           // tail: 250 = 7*32 + 26
            if (idx < CHUNKS) {
                v[7] = rp[idx];
            } else {
                const float ninf = -__builtin_inff();
                v[7] = make_float4(ninf, ninf, ninf, ninf);  // exp()->0, max() no-op
            }
        }

        // Pass 1: row max (registers only).
        float m = -__builtin_inff();
        #pragma unroll
        for (int k = 0; k < 8; ++k) {
            m = fmaxf(m, fmaxf(fmaxf(v[k].x, v[k].y), fmaxf(v[k].z, v[k].w)));
        }
        #pragma unroll
        for (int off = 16; off >= 1; off >>= 1) {
            m = fmaxf(m, __shfl_xor(m, off, 32));   // wave32 butterfly (5 steps)
        }

        // Pass 2: sum of exp(x - m) (registers only, v_exp_f32).
        float s = 0.0f;
        #pragma unroll
        for (int k = 0; k < 8; ++k) {
            s += __expf(v[k].x - m);
            s += __expf(v[k].y - m);
            s += __expf(v[k].z - m);
            s += __expf(v[k].w - m);
        }
        #pragma unroll
        for (int off = 16; off >= 1; off >>= 1) {
            s += __shfl_xor(s, off, 32);
        }

        if (lane == 0) {
            const float logZ  = m + __logf(s);              // v_log_f32 path
            const int   t     = tgt[row];
            const float xt    = inp[(size_t)row * ROW_C + t];  // L2/L0 hit: line just streamed
            const float logpt = xt - logZ;
            const float pt    = __expf(logpt);

            int b = (int)(pt * (float)NBINS);               // trunc == astype(int32)
            b = (b < 0) ? 0 : ((b > NBINS - 1) ? NBINS - 1 : b);
            const float g  = gammas[b];
            const float gs = (g > 0.0f) ? 1.0f : ((g < 0.0f) ? -1.0f : 0.0f);
            const float gm = fabsf(g);

            const float base = 1.0f - gs * pt + EPSF;       // > 0 always
            lossw = -__powf(base, gm) * logpt;
        }
    }

    if (lane == 0) wave_loss[wave] = lossw;
    __syncthreads();

    if (threadIdx.x == 0) {
        float total = 0.0f;
        #pragma unroll
        for (int w = 0; w < WAVES_PER_BLOCK; ++w) total += wave_loss[w];
        unsafeAtomicAdd(out, total);   // native global_atomic_add_f32, 1 per 8 rows
    }
}

__global__ void zero_out_kernel(float* out) {
    if (threadIdx.x == 0 && blockIdx.x == 0) out[0] = 0.0f;
}

extern "C" void kernel_launch(void* const* d_in, const int* in_sizes, int n_in,
                              void* d_out, int out_size, void* d_ws, size_t ws_size,
                              hipStream_t stream) {
    const float* inp    = (const float*)d_in[0];
    const int*   tgt    = (const int*)d_in[1];
    const float* gammas = (const float*)d_in[2];
    float*       out    = (float*)d_out;

    const int nrows = in_sizes[1];   // N = 65536

    zero_out_kernel<<<1, 32, 0, stream>>>(out);

    const int blocks = (nrows + WAVES_PER_BLOCK - 1) / WAVES_PER_BLOCK;
    adafocal_kernel<<<blocks, WAVES_PER_BLOCK * 32, 0, stream>>>(
        inp, tgt, gammas, out, nrows);
}